// QuantumTSTransformerRL_v5_83897891160625
// MI455X (gfx1250) — compile-verified
//
#include <hip/hip_runtime.h>

typedef __attribute__((ext_vector_type(2))) float v2f;
typedef __attribute__((ext_vector_type(8))) float v8f;

#define NQ      11
#define DIM     2048
#define BSZ     64
#define TSTEPS  8
#define NROTS   88
#define THREADS 256

// ---------------- scalar gate primitives on LDS statevector ----------------
// Flat index convention (JAX (2,)*n C-order): wire w <-> bit (10 - w).

__device__ __forceinline__ void gate_ry(float* sre, float* sim_, int p,
                                        float c, float s, int tid) {
  for (int j = tid; j < DIM / 2; j += THREADS) {
    int lo = j & ((1 << p) - 1);
    int i0 = ((j >> p) << (p + 1)) | lo;
    int i1 = i0 | (1 << p);
    float r0 = sre[i0], r1 = sre[i1], m0 = sim_[i0], m1 = sim_[i1];
    sre[i0] = c * r0 - s * r1;  sre[i1] = s * r0 + c * r1;
    sim_[i0] = c * m0 - s * m1; sim_[i1] = s * m0 + c * m1;
  }
}

// Controlled-RX: control bit pcb, target bit ptb; RX = [[c,-i s],[-i s,c]]
__device__ __forceinline__ void gate_crx(float* sre, float* sim_, int pcb, int ptb,
                                         float c, float s, int tid) {
  int pl = pcb < ptb ? pcb : ptb;
  int ph = pcb < ptb ? ptb : pcb;
  for (int j = tid; j < DIM / 4; j += THREADS) {
    int low = j & ((1 << pl) - 1);
    int mid = (j >> pl) & ((1 << (ph - 1 - pl)) - 1);
    int hi  = j >> (ph - 1);
    int idx = (hi << (ph + 1)) | (mid << (pl + 1)) | low;
    int i0 = idx | (1 << pcb);        // control = 1, target = 0
    int i1 = i0 | (1 << ptb);         // control = 1, target = 1
    float r0 = sre[i0], m0 = sim_[i0], r1 = sre[i1], m1 = sim_[i1];
    sre[i0] =  c * r0 + s * m1;  sim_[i0] =  c * m0 - s * r1;
    sre[i1] =  s * m0 + c * r1;  sim_[i1] = -s * r0 + c * m1;
  }
}

// ------------- RY sub-layer: 7 scalar gates + fused 4-qubit WMMA -----------
// Wires 7..10 (bits 3..0) act within 16-amp blocks: build G16 = kron of the
// four 2x2 RY matrices and apply S(128x16) <- S * G16^T on tensor cores,
// as a chain of 4 accumulating V_WMMA_F32_16X16X4_F32 ops (full f32).
__device__ __forceinline__ void ry_layer(float* sre, float* sim_, float* g16,
                                         const float* ch, const float* sh,
                                         int pb, int tid) {
  for (int w = 0; w < 7; ++w) {               // wires 0..6 -> bits 10..4
    gate_ry(sre, sim_, 10 - w, ch[pb + w], sh[pb + w], tid);
    __syncthreads();
  }
  {                                           // build G16[r][c] (RY is real)
    int r = tid >> 4, cc = tid & 15;
    float g = 1.0f;
#pragma unroll
    for (int bb = 0; bb < 4; ++bb) {          // bit bb <-> wire 10-bb
      int rb = (r >> bb) & 1, cb = (cc >> bb) & 1;
      float cv = ch[pb + 10 - bb], sv = sh[pb + 10 - bb];
      g *= (rb == cb) ? cv : (rb ? sv : -sv); // [[c,-s],[s,c]]
    }
    g16[tid] = g;
  }
  __syncthreads();
  {
    int lane = tid & 31, wid = tid >> 5;      // 8 wave32s
    int half = lane >> 4, mrow = lane & 15;
    int rowbase = wid * 16;                   // each wave owns rows [16w,16w+16)
#pragma unroll
    for (int plane = 0; plane < 2; ++plane) { // re, im (G16 real)
      float* S = plane ? sim_ : sre;
      v8f C = {};
#pragma unroll
      for (int kk = 0; kk < 4; ++kk) {        // K = 16 in 4 chunks of 4
        // A (16x4 f32): lane M = mrow, element p -> K = 2*half + p
        v2f A, B;
        A[0] = S[(rowbase + mrow) * 16 + kk * 4 + 2 * half + 0];
        A[1] = S[(rowbase + mrow) * 16 + kk * 4 + 2 * half + 1];
        // B (4x16 f32): B[K][N] = G16[N][K]; N = mrow, element p -> K = 2*half + p
        B[0] = g16[mrow * 16 + kk * 4 + 2 * half + 0];
        B[1] = g16[mrow * 16 + kk * 4 + 2 * half + 1];
        C = __builtin_amdgcn_wmma_f32_16x16x4_f32(false, A, false, B,
                                                  (short)0, C, false, false);
      }
      // D layout: VGPR r -> M = r + 8*half, N = mrow
#pragma unroll
      for (int r = 0; r < 8; ++r)
        S[(rowbase + r + 8 * half) * 16 + mrow] = C[r];
    }
  }
  __syncthreads();
}

// ------------------------------- kernels -----------------------------------

__global__ __launch_bounds__(THREADS)
void init_kernel(const float* __restrict__ base, const float* __restrict__ pc,
                 float* __restrict__ work, float* __restrict__ acc) {
  __shared__ float red[THREADS];
  int b = blockIdx.x, tid = threadIdx.x;
  const float* v = base + (size_t)b * DIM;
  float p = 0.0f;
  for (int i = tid; i < DIM; i += THREADS) { float x = v[i]; p += x * x; }
  red[tid] = p; __syncthreads();
  for (int s = THREADS / 2; s > 0; s >>= 1) {
    if (tid < s) red[tid] += red[tid + s];
    __syncthreads();
  }
  float inv = 1.0f / sqrtf(red[0]);
  float p0 = pc[0];
  for (int i = tid; i < DIM; i += THREADS) {
    float w = v[i] * inv;
    size_t o = ((size_t)b * DIM + i) * 2;
    work[o] = w;      work[o + 1] = 0.0f;
    acc[o]  = p0 * w; acc[o + 1]  = 0.0f;
  }
}

__global__ __launch_bounds__(THREADS)
void zero_kernel(float* p, int n) {
  int j = blockIdx.x * THREADS + threadIdx.x;
  if (j < n) p[j] = 0.0f;
}

__global__ __launch_bounds__(THREADS)
void evolve_kernel(const float* __restrict__ work, const float* __restrict__ up,
                   const float* __restrict__ lcu, float* __restrict__ dst, int det) {
  __shared__ float sre[DIM], sim_[DIM];
  __shared__ float ch[NROTS], sh[NROTS];
  __shared__ float g16[256];
  int b = blockIdx.x, t = blockIdx.y, tid = threadIdx.x;
  for (int i = tid; i < DIM; i += THREADS) {
    size_t o = ((size_t)b * DIM + i) * 2;
    sre[i] = work[o]; sim_[i] = work[o + 1];
  }
  const float* P = up + (size_t)(b * TSTEPS + t) * NROTS;
  if (tid < NROTS) { float h = 0.5f * P[tid]; ch[tid] = cosf(h); sh[tid] = sinf(h); }
  __syncthreads();

  for (int l = 0; l < 2; ++l) {
    int base = l * 44;
    ry_layer(sre, sim_, g16, ch, sh, base, tid);          // params base+0..10
    for (int i = 10; i >= 0; --i) {                       // CRX(i,(i+1)%11)
      int tw = (i + 1) % 11;
      gate_crx(sre, sim_, 10 - i, 10 - tw,
               ch[base + 11 + (10 - i)], sh[base + 11 + (10 - i)], tid);
      __syncthreads();
    }
    ry_layer(sre, sim_, g16, ch, sh, base + 22, tid);     // params base+22..32
    gate_crx(sre, sim_, 10 - 10, 10 - 9, ch[base + 33], sh[base + 33], tid);
    __syncthreads();
    for (int i = 0; i < 10; ++i) {                        // CRX(i,(i-1)%11)
      int tw = (i + 10) % 11;
      gate_crx(sre, sim_, 10 - i, 10 - tw,
               ch[base + 34 + i], sh[base + 34 + i], tid);
      __syncthreads();
    }
  }

  float cf = lcu[b * TSTEPS + t];
  if (det) {
    float* o = dst + (size_t)(b * TSTEPS + t) * DIM * 2;
    for (int i = tid; i < DIM; i += THREADS) {
      o[i * 2] = cf * sre[i]; o[i * 2 + 1] = cf * sim_[i];
    }
  } else {
    float* o = dst + (size_t)b * DIM * 2;
    for (int i = tid; i < DIM; i += THREADS) {
      atomicAdd(&o[i * 2], cf * sre[i]);
      atomicAdd(&o[i * 2 + 1], cf * sim_[i]);
    }
  }
}

__global__ __launch_bounds__(THREADS)
void combine_kernel(const float* __restrict__ src, const float* __restrict__ pc,
                    int k, float* __restrict__ work, float* __restrict__ acc, int det) {
  int j = blockIdx.x * THREADS + threadIdx.x;
  if (j >= BSZ * DIM * 2) return;
  int b = j >> 12, r = j & 4095;                // 4096 floats per statevector
  float sum;
  if (det) {
    sum = 0.0f;
#pragma unroll
    for (int t = 0; t < TSTEPS; ++t)
      sum += src[((size_t)(b * TSTEPS + t) << 12) + r];
  } else {
    sum = src[((size_t)b << 12) + r];
  }
  work[j] = sum;
  acc[j] += pc[k] * sum;
}

__global__ __launch_bounds__(THREADS)
void finalize_kernel(float* __restrict__ out, const float* __restrict__ pc) {
  int j = blockIdx.x * THREADS + threadIdx.x;
  if (j >= BSZ * DIM * 2) return;
  float z = fabsf(pc[0]) + fabsf(pc[1]) + fabsf(pc[2]) + fabsf(pc[3]);
  out[j] *= (1.0f / z);
}

// ------------------------------ launcher -----------------------------------

extern "C" void kernel_launch(void* const* d_in, const int* in_sizes, int n_in,
                              void* d_out, int out_size, void* d_ws, size_t ws_size,
                              hipStream_t stream) {
  const float* base = (const float*)d_in[0];   // (64,2048)  f32
  const float* up   = (const float*)d_in[1];   // (64,8,88)  f32
  const float* lcu  = (const float*)d_in[2];   // (64,8)     f32
  const float* pc   = (const float*)d_in[3];   // (4,)       f32
  float* acc  = (float*)d_out;                 // (64,2048,2) f32 — used as accumulator
  float* work = (float*)d_ws;                  // 64*4096 f32 = 1 MB

  const size_t WORK_F = (size_t)BSZ * DIM * 2;                 // 262144 floats
  const size_t EV_F   = (size_t)BSZ * TSTEPS * DIM * 2;        // 8 MB worth
  int det = (ws_size >= (WORK_F + EV_F) * sizeof(float)) ? 1 : 0;
  float* evbuf = (float*)((char*)d_ws + WORK_F * sizeof(float));

  const int NELEM = BSZ * DIM * 2;
  const int GRID1 = (NELEM + THREADS - 1) / THREADS;           // 1024 blocks

  init_kernel<<<BSZ, THREADS, 0, stream>>>(base, pc, work, acc);
  for (int k = 1; k <= 3; ++k) {
    if (!det)
      zero_kernel<<<GRID1, THREADS, 0, stream>>>(evbuf, NELEM);
    evolve_kernel<<<dim3(BSZ, TSTEPS), THREADS, 0, stream>>>(work, up, lcu, evbuf, det);
    combine_kernel<<<GRID1, THREADS, 0, stream>>>(evbuf, pc, k, work, acc, det);
  }
  finalize_kernel<<<GRID1, THREADS, 0, stream>>>(acc, pc);
}